// FullAttentionLayerStateful_90237262889425
// MI455X (gfx1250) — compile-verified
//
#include <hip/hip_runtime.h>
#include <math.h>

typedef __attribute__((ext_vector_type(2))) float v2f;
typedef __attribute__((ext_vector_type(8))) float v8f;

#define H    2048
#define II   8192
#define NH   16
#define NKV  4
#define HD   128
#define CC   8192
#define EPSV 1e-6f
#define SCALE 0.08838834764831845f   // 1/sqrt(128)

// flash-decode tiling
#define CHUNKS 4
#define POSB (CC / CHUNKS)     // 2048 positions per block
#define POSW (POSB / 8)        // 256 positions per wave
#define TILES (POSW / 16)      // 16 wmma tiles per wave

__device__ __forceinline__ float wave_sum(float v) {
#pragma unroll
  for (int o = 16; o > 0; o >>= 1) v += __shfl_xor(v, o, 32);
  return v;
}

// ---------------- RMSNorm (1 block, 256 thr) ----------------
__global__ void k_rmsnorm(const float* __restrict__ x, const float* __restrict__ w,
                          float* __restrict__ out) {
  __shared__ float red[8];
  int t = threadIdx.x;
  float s = 0.f;
  for (int i = t; i < H; i += 256) { float v = x[i]; s += v * v; }
  s = wave_sum(s);
  if ((t & 31) == 0) red[t >> 5] = s;
  __syncthreads();
  float tot = 0.f;
#pragma unroll
  for (int i = 0; i < 8; ++i) tot += red[i];
  float r = rsqrtf(tot / (float)H + EPSV);
  for (int i = t; i < H; i += 256) out[i] = x[i] * r * w[i];
}

// ---------------- fused QKV GEMV: 5120 rows, wave per row ----------------
__global__ void k_qkv(const float* __restrict__ Wq, const float* __restrict__ Wk,
                      const float* __restrict__ Wv, const float* __restrict__ h,
                      float* __restrict__ qg, float* __restrict__ kv,
                      float* __restrict__ vv) {
  __shared__ float hs[H];
  int t = threadIdx.x;
  for (int i = t; i < H; i += 256) hs[i] = h[i];
  __syncthreads();
  int wave = t >> 5, lane = t & 31;
  int row = blockIdx.x * 8 + wave;  // 0..5119
  const float* wr;
  float* out;
  int orow;
  if (row < 4096)      { wr = Wq + (size_t)row * H;          out = qg; orow = row; }
  else if (row < 4608) { wr = Wk + (size_t)(row - 4096) * H; out = kv; orow = row - 4096; }
  else                 { wr = Wv + (size_t)(row - 4608) * H; out = vv; orow = row - 4608; }
  float acc = 0.f;
#pragma unroll 4
  for (int j = 0; j < 16; ++j) {
    int idx = j * 128 + lane * 4;
    float4 w4 = *(const float4*)(wr + idx);
    float4 h4 = *(const float4*)(hs + idx);
    acc += w4.x * h4.x + w4.y * h4.y + w4.z * h4.z + w4.w * h4.w;
  }
  acc = wave_sum(acc);
  if (lane == 0) out[orow] = acc;
}

// ---------------- per-head RMSNorm + RoPE + gate sigmoid (1 block, 640 thr = 20 waves) ----------------
__global__ void k_qk_norm_rope(const float* __restrict__ qg, const float* __restrict__ kraw,
                               const float* __restrict__ qn, const float* __restrict__ knw,
                               const float* __restrict__ cosT, const float* __restrict__ sinT,
                               const int* __restrict__ pos,
                               float* __restrict__ qout, float* __restrict__ kout,
                               float* __restrict__ gout) {
  int t = threadIdx.x, wave = t >> 5, lane = t & 31;
  int p = pos[0];
  float c0 = cosT[p * 64 + lane],      s0 = sinT[p * 64 + lane];
  float c1 = cosT[p * 64 + lane + 32], s1 = sinT[p * 64 + lane + 32];
  if (wave < NH) {
    int hh = wave;
    float v0 = qg[hh * 256 + lane];
    float v1 = qg[hh * 256 + lane + 32];
    float v2 = qg[hh * 256 + lane + 64];
    float v3 = qg[hh * 256 + lane + 96];
    float ss = wave_sum(v0 * v0 + v1 * v1 + v2 * v2 + v3 * v3);
    float r = rsqrtf(ss / (float)HD + EPSV);
    float n0 = v0 * r * qn[lane];
    float n1 = v1 * r * qn[lane + 32];
    float n2 = v2 * r * qn[lane + 64];
    float n3 = v3 * r * qn[lane + 96];
    qout[hh * 128 + lane]      = n0 * c0 - n2 * s0;
    qout[hh * 128 + lane + 64] = n2 * c0 + n0 * s0;
    qout[hh * 128 + lane + 32] = n1 * c1 - n3 * s1;
    qout[hh * 128 + lane + 96] = n3 * c1 + n1 * s1;
#pragma unroll
    for (int j = 0; j < 4; ++j) {
      int d = lane + 32 * j;
      float g = qg[hh * 256 + 128 + d];
      gout[hh * 128 + d] = 1.f / (1.f + __expf(-g));
    }
  } else {
    int hh = wave - NH;  // 0..3
    float v0 = kraw[hh * 128 + lane];
    float v1 = kraw[hh * 128 + lane + 32];
    float v2 = kraw[hh * 128 + lane + 64];
    float v3 = kraw[hh * 128 + lane + 96];
    float ss = wave_sum(v0 * v0 + v1 * v1 + v2 * v2 + v3 * v3);
    float r = rsqrtf(ss / (float)HD + EPSV);
    float n0 = v0 * r * knw[lane];
    float n1 = v1 * r * knw[lane + 32];
    float n2 = v2 * r * knw[lane + 64];
    float n3 = v3 * r * knw[lane + 96];
    kout[hh * 128 + lane]      = n0 * c0 - n2 * s0;
    kout[hh * 128 + lane + 64] = n2 * c0 + n0 * s0;
    kout[hh * 128 + lane + 32] = n1 * c1 - n3 * s1;
    kout[hh * 128 + lane + 96] = n3 * c1 + n1 * s1;
  }
}

// ---------------- flash-decode pass 1 with WMMA scores ----------------
// grid (NKV, CHUNKS), 256 threads. Scores via V_WMMA_F32_16X16X4_F32:
// M = 16 cache positions, N = 16 (heads, 4 valid), K stepped 4 over HD=128.
// A fragment (16x4 f32): lanes 0-15 hold K=0,1 ; lanes 16-31 hold K=2,3 ; M = lane%16.
// B fragment (4x16 f32): mirrored, staged in LDS once per block.
__global__ void k_attn1(const float* __restrict__ cache, const float* __restrict__ q,
                        const float* __restrict__ knew, const float* __restrict__ vnew,
                        const float* __restrict__ mask, const int* __restrict__ pos,
                        float* __restrict__ pm, float* __restrict__ pl,
                        float* __restrict__ pacc) {
  __shared__ float Bq[32][64];       // [kk][lane*2+t]
  __shared__ float stile[8][16][4];  // per-wave score tile (16 pos x 4 heads)
  int g = blockIdx.x, chunk = blockIdx.y;
  int t = threadIdx.x, wave = t >> 5, lane = t & 31;
  int p = pos[0];
  const float* Kbase = cache + (size_t)g * CC * HD;
  const float* Vbase = cache + (size_t)(NKV + g) * CC * HD;

  // stage B = q^T fragments for this kv-group (heads g*4..g*4+3 in N=0..3, rest zero)
  for (int e = t; e < 32 * 64; e += 256) {
    int kk = e >> 6, i = e & 63;
    int ln = i >> 1, tt = i & 1;
    int nn = ln & 15, k0 = 2 * (ln >> 4);
    float val = 0.f;
    if (nn < 4) val = q[(g * 4 + nn) * HD + kk * 4 + k0 + tt];
    Bq[kk][i] = val;
  }
  __syncthreads();

  int n = lane & 15, half = lane >> 4;
  float mh[4], lh[4], acc[4][4];
#pragma unroll
  for (int h = 0; h < 4; ++h) {
    mh[h] = -1e30f; lh[h] = 0.f;
    for (int j = 0; j < 4; ++j) acc[h][j] = 0.f;
  }

  int cw = chunk * POSB + wave * POSW;
  for (int tl = 0; tl < TILES; ++tl) {
    int c0 = cw + tl * 16;
    int crow = c0 + n;  // this lane's A-matrix row (cache position)
    const float* arow = (crow == p) ? (knew + g * HD) : (Kbase + (size_t)crow * HD);
    v8f D = {0.f, 0.f, 0.f, 0.f, 0.f, 0.f, 0.f, 0.f};
#pragma unroll
    for (int kk = 0; kk < 32; ++kk) {
      v2f a; a.x = arow[kk * 4 + 2 * half]; a.y = arow[kk * 4 + 2 * half + 1];
      v2f b; b.x = Bq[kk][lane * 2];        b.y = Bq[kk][lane * 2 + 1];
      D = __builtin_amdgcn_wmma_f32_16x16x4_f32(false, a, false, b, (short)0, D,
                                                false, false);
    }
    // D vgpr j holds S[m = j + 8*half][n = lane&15]
    if (n < 4) {
#pragma unroll
      for (int j = 0; j < 8; ++j) {
        int m = j + 8 * half;
        stile[wave][m][n] = D[j] * SCALE + mask[c0 + m];
      }
    }
    __syncthreads();

    // online-softmax update (all lanes hold replicated per-head scalars)
#pragma unroll
    for (int h = 0; h < 4; ++h) {
      float mx = -1e30f;
      for (int m = 0; m < 16; ++m) mx = fmaxf(mx, stile[wave][m][h]);
      float nm = fmaxf(mh[h], mx);
      float r = __expf(mh[h] - nm);
      lh[h] *= r;
      for (int j = 0; j < 4; ++j) acc[h][j] *= r;
      mh[h] = nm;
    }
    for (int m = 0; m < 16; ++m) {
      int c = c0 + m;
      const float* vrow = (c == p) ? (vnew + g * HD) : (Vbase + (size_t)c * HD);
      float vv0 = vrow[lane], vv1 = vrow[lane + 32];
      float vv2 = vrow[lane + 64], vv3 = vrow[lane + 96];
#pragma unroll
      for (int h = 0; h < 4; ++h) {
        float w = __expf(stile[wave][m][h] - mh[h]);
        lh[h] += w;
        acc[h][0] += w * vv0; acc[h][1] += w * vv1;
        acc[h][2] += w * vv2; acc[h][3] += w * vv3;
      }
    }
    __syncthreads();
  }

  int tix = chunk * 8 + wave;  // 0..31 partial slot
#pragma unroll
  for (int h = 0; h < 4; ++h) {
    int qh = g * 4 + h;
    if (lane == 0) { pm[qh * 32 + tix] = mh[h]; pl[qh * 32 + tix] = lh[h]; }
    for (int j = 0; j < 4; ++j)
      pacc[((size_t)(qh * 32 + tix)) * HD + lane + 32 * j] = acc[h][j];
  }
}

// ---------------- flash-decode pass 2: combine partials + sigmoid gate ----------------
__global__ void k_attn2(const float* __restrict__ pm, const float* __restrict__ pl,
                        const float* __restrict__ pacc, const float* __restrict__ gate,
                        float* __restrict__ out) {
  int hq = blockIdx.x, d = threadIdx.x;
  float M = -1e30f;
  for (int t = 0; t < 32; ++t) M = fmaxf(M, pm[hq * 32 + t]);
  float L = 0.f, s = 0.f;
  for (int t = 0; t < 32; ++t) {
    float w = __expf(pm[hq * 32 + t] - M);
    L += w * pl[hq * 32 + t];
    s += w * pacc[((size_t)(hq * 32 + t)) * HD + d];
  }
  out[hq * HD + d] = (s / L) * gate[hq * HD + d];
}

// ---------------- Wo GEMV + residual ----------------
__global__ void k_o_proj(const float* __restrict__ Wo, const float* __restrict__ attn,
                         const float* __restrict__ resid, float* __restrict__ x) {
  __shared__ float as[H];
  int t = threadIdx.x;
  for (int i = t; i < H; i += 256) as[i] = attn[i];
  __syncthreads();
  int wave = t >> 5, lane = t & 31;
  int row = blockIdx.x * 8 + wave;
  const float* wr = Wo + (size_t)row * H;
  float acc = 0.f;
#pragma unroll 4
  for (int j = 0; j < 16; ++j) {
    int idx = j * 128 + lane * 4;
    float4 w4 = *(const float4*)(wr + idx);
    float4 a4 = *(const float4*)(as + idx);
    acc += w4.x * a4.x + w4.y * a4.y + w4.z * a4.z + w4.w * a4.w;
  }
  acc = wave_sum(acc);
  if (lane == 0) x[row] = resid[row] + acc;
}

// ---------------- gate/up GEMV + SiLU ----------------
__global__ void k_gateup(const float* __restrict__ Wg, const float* __restrict__ Wu,
                         const float* __restrict__ h2, float* __restrict__ gu) {
  __shared__ float hs[H];
  int t = threadIdx.x;
  for (int i = t; i < H; i += 256) hs[i] = h2[i];
  __syncthreads();
  int wave = t >> 5, lane = t & 31;
  int row = blockIdx.x * 8 + wave;
  const float* gr = Wg + (size_t)row * H;
  const float* ur = Wu + (size_t)row * H;
  float ag = 0.f, au = 0.f;
#pragma unroll 4
  for (int j = 0; j < 16; ++j) {
    int idx = j * 128 + lane * 4;
    float4 g4 = *(const float4*)(gr + idx);
    float4 u4 = *(const float4*)(ur + idx);
    float4 h4 = *(const float4*)(hs + idx);
    ag += g4.x * h4.x + g4.y * h4.y + g4.z * h4.z + g4.w * h4.w;
    au += u4.x * h4.x + u4.y * h4.y + u4.z * h4.z + u4.w * h4.w;
  }
  ag = wave_sum(ag); au = wave_sum(au);
  if (lane == 0) {
    float s = ag / (1.f + __expf(-ag));  // silu
    gu[row] = s * au;
  }
}

// ---------------- down GEMV + residual -> final output ----------------
__global__ void k_down(const float* __restrict__ Wd, const float* __restrict__ gu,
                       const float* __restrict__ x, float* __restrict__ out) {
  __shared__ float gs[II];  // 32 KB
  int t = threadIdx.x;
  for (int i = t; i < II; i += 256) gs[i] = gu[i];
  __syncthreads();
  int wave = t >> 5, lane = t & 31;
  int row = blockIdx.x * 8 + wave;
  const float* wr = Wd + (size_t)row * II;
  float acc = 0.f;
#pragma unroll 4
  for (int j = 0; j < 64; ++j) {
    int idx = j * 128 + lane * 4;
    float4 w4 = *(const float4*)(wr + idx);
    float4 g4 = *(const float4*)(gs + idx);
    acc += w4.x * g4.x + w4.y * g4.y + w4.z * g4.z + w4.w * g4.w;
  }
  acc = wave_sum(acc);
  if (lane == 0) out[row] = x[row] + acc;
}

extern "C" void kernel_launch(void* const* d_in, const int* in_sizes, int n_in,
                              void* d_out, int out_size, void* d_ws, size_t ws_size,
                              hipStream_t stream) {
  const float* hidden = (const float*)d_in[0];
  const int*   pos    = (const int*)d_in[1];
  const float* mask   = (const float*)d_in[2];
  const float* Wq     = (const float*)d_in[3];
  const float* Wk     = (const float*)d_in[4];
  const float* Wv     = (const float*)d_in[5];
  const float* Wo     = (const float*)d_in[6];
  const float* q_norm = (const float*)d_in[7];
  const float* k_norm = (const float*)d_in[8];
  const float* ln1_w  = (const float*)d_in[9];
  const float* ln2_w  = (const float*)d_in[10];
  const float* Wg     = (const float*)d_in[11];
  const float* Wu     = (const float*)d_in[12];
  const float* Wd     = (const float*)d_in[13];
  const float* cache  = (const float*)d_in[14];
  const float* rcos   = (const float*)d_in[15];
  const float* rsin   = (const float*)d_in[16];
  float* out = (float*)d_out;

  float* ws = (float*)d_ws;
  float* ws_h       = ws;            // 2048
  float* ws_qg      = ws + 2048;     // 4096
  float* ws_k       = ws + 6144;     // 512
  float* ws_v       = ws + 6656;     // 512
  float* ws_q       = ws + 7168;     // 2048
  float* ws_kn      = ws + 9216;     // 512
  float* ws_gate    = ws + 9728;     // 2048
  float* ws_x       = ws + 11776;    // 2048
  float* ws_h2      = ws + 13824;    // 2048
  float* ws_gu      = ws + 15872;    // 8192
  float* ws_attnout = ws + 24064;    // 2048
  float* ws_pm      = ws + 26112;    // 512  (16 heads x 32 partials)
  float* ws_pl      = ws + 26624;    // 512
  float* ws_pacc    = ws + 27136;    // 65536 (16*32*128)

  k_rmsnorm<<<1, 256, 0, stream>>>(hidden, ln1_w, ws_h);
  k_qkv<<<640, 256, 0, stream>>>(Wq, Wk, Wv, ws_h, ws_qg, ws_k, ws_v);
  k_qk_norm_rope<<<1, 640, 0, stream>>>(ws_qg, ws_k, q_norm, k_norm, rcos, rsin,
                                        pos, ws_q, ws_kn, ws_gate);
  k_attn1<<<dim3(NKV, CHUNKS), 256, 0, stream>>>(cache, ws_q, ws_kn, ws_v, mask,
                                                 pos, ws_pm, ws_pl, ws_pacc);
  k_attn2<<<NH, HD, 0, stream>>>(ws_pm, ws_pl, ws_pacc, ws_gate, ws_attnout);
  k_o_proj<<<256, 256, 0, stream>>>(Wo, ws_attnout, hidden, ws_x);
  k_rmsnorm<<<1, 256, 0, stream>>>(ws_x, ln2_w, ws_h2);
  k_gateup<<<1024, 256, 0, stream>>>(Wg, Wu, ws_h2, ws_gu);
  k_down<<<256, 256, 0, stream>>>(Wd, ws_gu, ws_x, out);
}